// Image2PointGridSample_62234076119233
// MI455X (gfx1250) — compile-verified
//
#include <hip/hip_runtime.h>
#include <hip/hip_bf16.h>
#include <stddef.h>

#define NPTS   200000
#define BATCH  4
#define CVOX   64
#define CIMG   256
#define HFEA   48
#define WFEA   160
#define OUTC   (CVOX + CIMG)   // 320
#define PPB    128             // points per block = 8 waves * 16

typedef __attribute__((ext_vector_type(2))) float v2f;
typedef __attribute__((ext_vector_type(8))) float v8f;
typedef __attribute__((ext_vector_type(2))) float f32x2;
typedef __attribute__((ext_vector_type(4))) float f32x4;

#ifndef __has_builtin
#define __has_builtin(x) 0
#endif
#if defined(__AMDGCN__) && __has_builtin(__builtin_amdgcn_wmma_f32_16x16x4_f32)
#define USE_WMMA 1
#else
#define USE_WMMA 0
#endif

__device__ __forceinline__ float fastrcp(float x) {
#if defined(__AMDGCN__) && __has_builtin(__builtin_amdgcn_rcpf)
    return __builtin_amdgcn_rcpf(x);     // v_rcp_f32
#else
    return 1.0f / x;
#endif
}

__device__ __forceinline__ float lane_bcast(float v, int srcLane) {
#if defined(__AMDGCN__) && __has_builtin(__builtin_amdgcn_readlane)
    return __int_as_float(__builtin_amdgcn_readlane(__float_as_int(v), srcLane));
#else
    return __shfl(v, srcLane, 32);
#endif
}

// ---------------------------------------------------------------------------
// Pass 1: transpose img_feats [B,CI,H,W] -> ws [B,H,W,CI] so per-corner
// channel gathers in the main kernel are contiguous 1KB reads (full L2 lines).
// 32x32 LDS tile; P = H*W = 7680 (=240*32), CI = 256 (=8*32).
// ---------------------------------------------------------------------------
__global__ __launch_bounds__(256) void transpose_feats(const float* __restrict__ in,
                                                       float* __restrict__ out) {
    __shared__ float tile[32][33];
    const int P = HFEA * WFEA;
    const int bx = blockIdx.x;           // P/32 tiles
    const int by = blockIdx.y;           // CI/32 tiles
    const int bz = blockIdx.z;           // batch
    const int tx = threadIdx.x;          // 0..31
    const int ty = threadIdx.y;          // 0..7

    const float* src = in + (size_t)bz * CIMG * P;
#pragma unroll
    for (int k = 0; k < 4; ++k) {
        int c = by * 32 + ty + k * 8;
        int p = bx * 32 + tx;
        tile[ty + k * 8][tx] = src[(size_t)c * P + p];   // coalesced in p
    }
    __syncthreads();
    float* dst = out + (size_t)bz * P * CIMG;
#pragma unroll
    for (int k = 0; k < 4; ++k) {
        int p = bx * 32 + ty + k * 8;
        int c = by * 32 + tx;
        dst[(size_t)p * CIMG + c] = tile[tx][ty + k * 8]; // coalesced in c
    }
}

// ---------------------------------------------------------------------------
// Main kernel: each wave32 handles 16 points.
//  * Projection of 16 points = (16x4 xyz1) x (4x16 M^T) via one
//    V_WMMA_F32_16X16X4_F32 (batch uniform per 16-pt group: 50000%16==0).
//  * Per point: bilinear gather, whole wave on one point, lane = 8 channels.
// ---------------------------------------------------------------------------
template <int TRANSPOSED>
__global__ __launch_bounds__(256) void sample_kernel(
    const float* __restrict__ pm,      // points_mean  N*3
    const float* __restrict__ vfeat,   // voxel_feats  N*64
    const int*   __restrict__ coors,   // N*4 (col 0 = batch)
    const float* __restrict__ imgf,    // [B,CI,H,W] original
    const float* __restrict__ ftr,     // [B,H,W,CI] transposed (ws) or null
    const float* __restrict__ l2i,     // B*16
    const int*   __restrict__ ps,      // B*2 pad_shape
    float* __restrict__ out)           // N*320
{
    const int lane = threadIdx.x & 31;
    const int wave = threadIdx.x >> 5;
    const int base = blockIdx.x * PPB + wave * 16;
    if (base >= NPTS) return;

    const int bb = coors[4 * base];                 // uniform per wave
    // ix = x_pix/padW*(W-1), iy = y_pix/padH*(H-1): fold into one scale each
    const float sclX = (float)(WFEA - 1) * fastrcp((float)ps[2 * bb + 1]);
    const float sclY = (float)(HFEA - 1) * fastrcp((float)ps[2 * bb + 0]);

#if USE_WMMA
    // A (16x4 f32): lanes 0-15 hold K=0,1 (x,y); lanes 16-31 hold K=2,3 (z,1).
    // B (4x16): B[k,n] = M[n,k]; VGPR0 = K rows 0|2, VGPR1 = K rows 1|3.
    v8f d;
    {
        const int  m  = lane & 15;
        const bool hi = lane >= 16;
        const float* pp = pm + 3 * (size_t)(base + m);
        v2f A;
        if (hi) { A.x = pp[2]; A.y = 1.0f; }
        else    { A.x = pp[0]; A.y = pp[1]; }
        v2f B;
        const int nn = lane & 15;
        if (nn < 4) {
            const int k0 = hi ? 2 : 0;
            B.x = l2i[bb * 16 + nn * 4 + k0];
            B.y = l2i[bb * 16 + nn * 4 + k0 + 1];
        } else {
            B.x = 0.0f; B.y = 0.0f;
        }
        v8f C = {0.f, 0.f, 0.f, 0.f, 0.f, 0.f, 0.f, 0.f};
        d = __builtin_amdgcn_wmma_f32_16x16x4_f32(false, A, false, B,
                                                  (short)0, C, false, false);
    }
#endif

#pragma unroll
    for (int p = 0; p < 16; ++p) {
        const int n = base + p;
        float X, Y, Z;
#if USE_WMMA
        // D layout: VGPR r -> rows r (lanes 0-15) / r+8 (lanes 16-31), N=lane%16
        const int src = ((p >> 3) << 4);
        X = lane_bcast(d[p & 7], src + 0);   // v_readlane_b32 (uniform)
        Y = lane_bcast(d[p & 7], src + 1);
        Z = lane_bcast(d[p & 7], src + 2);
#else
        const float px = pm[3 * (size_t)n + 0];
        const float py = pm[3 * (size_t)n + 1];
        const float pz = pm[3 * (size_t)n + 2];
        const float* M = l2i + bb * 16;
        X = M[0] * px + M[1] * py + M[2]  * pz + M[3];
        Y = M[4] * px + M[5] * py + M[6]  * pz + M[7];
        Z = M[8] * px + M[9] * py + M[10] * pz + M[11];
#endif
        const float rz = fastrcp(fmaxf(Z, 1e-5f));   // v_rcp_f32, no div chain
        const float ix = X * rz * sclX;
        const float iy = Y * rz * sclY;
        const float ix0 = floorf(ix), iy0 = floorf(iy);
        const float wx1 = ix - ix0, wx0 = 1.0f - wx1;
        const float wy1 = iy - iy0, wy0 = 1.0f - wy1;

        float acc[8];
#pragma unroll
        for (int j = 0; j < 8; ++j) acc[j] = 0.0f;

        auto corner = [&](float yf, float xf, float w) {
            // All operands wave-uniform -> uniform branch skips OOB / zero-w
            if (w != 0.0f && yf >= 0.0f && yf < (float)HFEA &&
                xf >= 0.0f && xf < (float)WFEA) {
                const int yi = (int)yf, xi = (int)xf;
                if (TRANSPOSED) {
                    const f32x4* s = (const f32x4*)(ftr +
                        (((size_t)bb * HFEA + yi) * WFEA + xi) * CIMG + lane * 8);
                    const f32x4 v0 = s[0], v1 = s[1];
                    acc[0] += w * v0.x; acc[1] += w * v0.y;
                    acc[2] += w * v0.z; acc[3] += w * v0.w;
                    acc[4] += w * v1.x; acc[5] += w * v1.y;
                    acc[6] += w * v1.z; acc[7] += w * v1.w;
                } else {
                    const size_t sb = (size_t)bb * CIMG * HFEA * WFEA +
                                      (size_t)yi * WFEA + xi;
                    const int c0 = lane * 8;
#pragma unroll
                    for (int j = 0; j < 8; ++j)
                        acc[j] += w * imgf[sb + (size_t)(c0 + j) * (HFEA * WFEA)];
                }
            }
        };
        corner(iy0,       ix0,       wy0 * wx0);
        corner(iy0,       ix0 + 1.f, wy0 * wx1);
        corner(iy0 + 1.f, ix0,       wy1 * wx0);
        corner(iy0 + 1.f, ix0 + 1.f, wy1 * wx1);

        // voxel_feats passthrough: 64 floats, 2 per lane; streamed (NT)
        {
            const f32x2* vs = (const f32x2*)(vfeat + (size_t)n * CVOX);
            f32x2*       vd = (f32x2*)(out + (size_t)n * OUTC);
            f32x2 v = __builtin_nontemporal_load(vs + lane);
            __builtin_nontemporal_store(v, vd + lane);
        }
        // point feats: 8 floats per lane as two b128 NT stores (write-once)
        {
            f32x4 o0 = {acc[0], acc[1], acc[2], acc[3]};
            f32x4 o1 = {acc[4], acc[5], acc[6], acc[7]};
            f32x4* pd = (f32x4*)(out + (size_t)n * OUTC + CVOX);
            __builtin_nontemporal_store(o0, pd + lane * 2 + 0);
            __builtin_nontemporal_store(o1, pd + lane * 2 + 1);
        }
    }
}

extern "C" void kernel_launch(void* const* d_in, const int* in_sizes, int n_in,
                              void* d_out, int out_size, void* d_ws, size_t ws_size,
                              hipStream_t stream) {
    const float* pm    = (const float*)d_in[0];  // points_mean (N,1,3)
    /* d_in[1] = mask (unused) */
    const float* vfeat = (const float*)d_in[2];  // voxel_feats (N,64)
    const int*   coors = (const int*)  d_in[3];  // (N,4)
    const float* imgf  = (const float*)d_in[4];  // (B,CI,H,W)
    const float* l2i   = (const float*)d_in[5];  // (B,4,4)
    const int*   ps    = (const int*)  d_in[6];  // (B,2)
    float* out = (float*)d_out;

    const size_t need = (size_t)BATCH * CIMG * HFEA * WFEA * sizeof(float);
    const int blocks = (NPTS + PPB - 1) / PPB;

    if (d_ws != nullptr && ws_size >= need) {
        float* ftr = (float*)d_ws;
        dim3 tg((HFEA * WFEA) / 32, CIMG / 32, BATCH);
        transpose_feats<<<tg, dim3(32, 8, 1), 0, stream>>>(imgf, ftr);
        sample_kernel<1><<<blocks, 256, 0, stream>>>(pm, vfeat, coors, imgf, ftr,
                                                     l2i, ps, out);
    } else {
        sample_kernel<0><<<blocks, 256, 0, stream>>>(pm, vfeat, coors, imgf,
                                                     nullptr, l2i, ps, out);
    }
}